// DeepGCN_70085276336554
// MI455X (gfx1250) — compile-verified
//
#include <hip/hip_runtime.h>

#define N_NODES   5000
#define N_EDGES   50000
#define HID       64
#define OUTC      12
#define SP        48            // H*W = 12*4
#define CHW       (HID * SP)    // 3072
#define NODE_ELEMS (N_NODES * CHW)   // 15,360,000
#define EDGE_ELEMS (N_EDGES * CHW)   // 153,600,000
#define NODE_Q    (NODE_ELEMS / 4)   // 3,840,000 float4 groups
#define EDGE_Q    (N_EDGES * (CHW / 4))  // 38,400,000 float4 groups

typedef __attribute__((ext_vector_type(2))) float v2f;
typedef __attribute__((ext_vector_type(8))) float v8f;

// ---------------------------------------------------------------- utilities
__global__ void copy4_kernel(float4* __restrict__ d, const float4* __restrict__ s, int n4) {
    int i = blockIdx.x * blockDim.x + threadIdx.x;
    if (i < n4) d[i] = s[i];
}

__global__ void zero4_kernel(float4* __restrict__ p, int n4) {
    int i = blockIdx.x * blockDim.x + threadIdx.x;
    int stride = gridDim.x * blockDim.x;
    float4 z = make_float4(0.f, 0.f, 0.f, 0.f);
    for (; i < n4; i += stride) p[i] = z;
}

// ---------------------------------------------------------------- batchnorm
// Per-channel sum / sum-of-squares. grid = (64 channels, 20 node-chunks)
__global__ void bn_stats(const float* __restrict__ hv, float* __restrict__ sums,
                         float* __restrict__ sqs) {
    __shared__ float sh_s[256];
    __shared__ float sh_q[256];
    const int c = blockIdx.x;
    const int n0 = blockIdx.y * (N_NODES / 20);   // 250 nodes per chunk
    float s = 0.0f, q = 0.0f;
    for (int t = threadIdx.x; t < (N_NODES / 20) * (SP / 4); t += blockDim.x) {
        int n = n0 + t / (SP / 4);
        int quad = t % (SP / 4);
        const float4 v = *(const float4*)(hv + n * CHW + c * SP + quad * 4);
        s += v.x + v.y + v.z + v.w;
        q += v.x * v.x + v.y * v.y + v.z * v.z + v.w * v.w;
    }
    sh_s[threadIdx.x] = s;
    sh_q[threadIdx.x] = q;
    __syncthreads();
    for (int off = 128; off > 0; off >>= 1) {
        if (threadIdx.x < off) {
            sh_s[threadIdx.x] += sh_s[threadIdx.x + off];
            sh_q[threadIdx.x] += sh_q[threadIdx.x + off];
        }
        __syncthreads();
    }
    if (threadIdx.x == 0) {
        atomicAdd(&sums[c], sh_s[0]);
        atomicAdd(&sqs[c], sh_q[0]);
    }
}

// stats[0..63]=sums, [64..127]=sqs -> write coef=[128..191], off=[192..255]
__global__ void bn_final(float* __restrict__ stats, const float* __restrict__ gamma,
                         const float* __restrict__ beta) {
    int c = threadIdx.x;      // 64 threads
    const float inv_n = 1.0f / (float)(N_NODES * SP);
    float mean = stats[c] * inv_n;
    float var = stats[64 + c] * inv_n - mean * mean;
    float rs = rsqrtf(var + 1e-5f);
    float coef = gamma[c] * rs;
    stats[128 + c] = coef;
    stats[192 + c] = beta[c] - mean * coef;
}

// hv1 = relu(hv * coef[c] + off[c]) -- one float4 per thread, c uniform in quad
__global__ void bn_apply(const float* __restrict__ hv, float* __restrict__ hv1,
                         const float* __restrict__ coef, const float* __restrict__ off) {
    int i = blockIdx.x * blockDim.x + threadIdx.x;
    if (i >= NODE_Q) return;
    int c = (i / (SP / 4)) & (HID - 1);
    const float co = coef[c], of = off[c];
    const float4 v = *(const float4*)(hv + i * 4);
    float4 r;
    r.x = fmaxf(v.x * co + of, 0.f);
    r.y = fmaxf(v.y * co + of, 0.f);
    r.z = fmaxf(v.z * co + of, 0.f);
    r.w = fmaxf(v.w * co + of, 0.f);
    *(float4*)(hv1 + i * 4) = r;
}

// ---------------------------------------------------------------- edge passes
// pass 1: segment max of m = relu(hv1[src]+he)+eps (> 0 -> int atomicMax on zeros)
__global__ void edge_pass1(const float* __restrict__ hv1, const float* __restrict__ he,
                           const int* __restrict__ src, const int* __restrict__ dst,
                           float* __restrict__ mx) {
    int i = blockIdx.x * blockDim.x + threadIdx.x;
    if (i >= EDGE_Q) return;
    int e = i / (CHW / 4), r = (i % (CHW / 4)) * 4;
    const float4 hs = *(const float4*)(hv1 + src[e] * CHW + r);
    const float4 hh = *(const float4*)(he + e * CHW + r);
    int* mb = (int*)(mx + dst[e] * CHW + r);
    atomicMax(mb + 0, __float_as_int(fmaxf(hs.x + hh.x, 0.f) + 1e-7f));
    atomicMax(mb + 1, __float_as_int(fmaxf(hs.y + hh.y, 0.f) + 1e-7f));
    atomicMax(mb + 2, __float_as_int(fmaxf(hs.z + hh.z, 0.f) + 1e-7f));
    atomicMax(mb + 3, __float_as_int(fmaxf(hs.w + hh.w, 0.f) + 1e-7f));
}

// pass 2: ssum[dst] += exp(m - mx[dst]);  agg[dst] += exp(m - mx[dst]) * m
// (softmax-weighted sum folded: Sum(a*m) = Sum(e*m)/Sum(e))
__global__ void edge_pass2(const float* __restrict__ hv1, const float* __restrict__ he,
                           const int* __restrict__ src, const int* __restrict__ dst,
                           const float* __restrict__ mx, float* __restrict__ ssum,
                           float* __restrict__ agg) {
    int i = blockIdx.x * blockDim.x + threadIdx.x;
    if (i >= EDGE_Q) return;
    int e = i / (CHW / 4), r = (i % (CHW / 4)) * 4;
    const float4 hs = *(const float4*)(hv1 + src[e] * CHW + r);
    const float4 hh = *(const float4*)(he + e * CHW + r);
    const int di = dst[e] * CHW + r;
    const float4 mv = *(const float4*)(mx + di);
    float m0 = fmaxf(hs.x + hh.x, 0.f) + 1e-7f;
    float m1 = fmaxf(hs.y + hh.y, 0.f) + 1e-7f;
    float m2 = fmaxf(hs.z + hh.z, 0.f) + 1e-7f;
    float m3 = fmaxf(hs.w + hh.w, 0.f) + 1e-7f;
    float e0 = __expf(m0 - mv.x), e1 = __expf(m1 - mv.y);
    float e2 = __expf(m2 - mv.z), e3 = __expf(m3 - mv.w);
    atomicAdd(&ssum[di + 0], e0);
    atomicAdd(&ssum[di + 1], e1);
    atomicAdd(&ssum[di + 2], e2);
    atomicAdd(&ssum[di + 3], e3);
    atomicAdd(&agg[di + 0], e0 * m0);
    atomicAdd(&agg[di + 1], e1 * m1);
    atomicAdd(&agg[di + 2], e2 * m2);
    atomicAdd(&agg[di + 3], e3 * m3);
}

// feats = hv1 + agg/ssum (0 for edge-less nodes); written in place into agg
__global__ void feats_kernel(float* __restrict__ agg, const float* __restrict__ ssum,
                             const float* __restrict__ hv1) {
    int i = blockIdx.x * blockDim.x + threadIdx.x;
    if (i >= NODE_Q) return;
    const float4 a = *(const float4*)(agg + i * 4);
    const float4 s = *(const float4*)(ssum + i * 4);
    const float4 h = *(const float4*)(hv1 + i * 4);
    float4 r;
    r.x = h.x + (s.x > 0.f ? a.x / s.x : 0.f);
    r.y = h.y + (s.y > 0.f ? a.y / s.y : 0.f);
    r.z = h.z + (s.z > 0.f ? a.z / s.z : 0.f);
    r.w = h.w + (s.w > 0.f ? a.w / s.w : 0.f);
    *(float4*)(agg + i * 4) = r;
}

// ---------------------------------------------------------------- WMMA GEMMs
// hv[n,o,hw] += sum_c feats[n,c,hw]*W[c,o] + b[o]   (residual in-place)
// One wave computes a 16x64 strip: A fragment shared across 4 N-tiles.
__global__ __launch_bounds__(256) void genconv_gemm(const float* __restrict__ feats,
                                                    const float* __restrict__ W,
                                                    const float* __restrict__ b,
                                                    float* __restrict__ hv) {
    const int lane = threadIdx.x & 31;
    const int mt = (blockIdx.x * blockDim.x + threadIdx.x) >> 5;  // 15000 waves
    const int r0 = mt * 16;
    const int node = r0 / SP;
    const int hw0 = r0 % SP;          // 0/16/32 -> tile stays inside one node
    const int am = lane & 15;         // M (A) / N (B) index within tile
    const int ah = lane >> 4;         // selects K pair {0,1} vs {2,3}
    const float* fb = feats + node * CHW + hw0;
    v8f c0 = {}, c1 = {}, c2 = {}, c3 = {};
#pragma unroll
    for (int k0 = 0; k0 < HID; k0 += 4) {
        const int ak = k0 + ah * 2;
        v2f a, b0, b1, b2, b3;
        a.x = fb[ak * SP + am];                 // A[M=am][K=ak]
        a.y = fb[(ak + 1) * SP + am];           // A[M=am][K=ak+1]
        const float* w0 = W + ak * HID + am;
        const float* w1 = W + (ak + 1) * HID + am;
        b0.x = w0[0];  b0.y = w1[0];
        b1.x = w0[16]; b1.y = w1[16];
        b2.x = w0[32]; b2.y = w1[32];
        b3.x = w0[48]; b3.y = w1[48];
        c0 = __builtin_amdgcn_wmma_f32_16x16x4_f32(false, a, false, b0, (short)0, c0, false, false);
        c1 = __builtin_amdgcn_wmma_f32_16x16x4_f32(false, a, false, b1, (short)0, c1, false, false);
        c2 = __builtin_amdgcn_wmma_f32_16x16x4_f32(false, a, false, b2, (short)0, c2, false, false);
        c3 = __builtin_amdgcn_wmma_f32_16x16x4_f32(false, a, false, b3, (short)0, c3, false, false);
    }
    // D layout: VGPR j -> M = j + 8*ah, N = am
#pragma unroll
    for (int nt = 0; nt < 4; ++nt) {
        const v8f& c = (nt == 0) ? c0 : (nt == 1) ? c1 : (nt == 2) ? c2 : c3;
        const int o = nt * 16 + am;
        const float bo = b[o];
        float* ob = hv + node * CHW + o * SP + hw0 + ah * 8;
#pragma unroll
        for (int j = 0; j < 8; ++j) ob[j] += c[j] + bo;
    }
}

// out[n,o,hw] = sum_c hg[n,c,hw]*W[c,o] + b[o],  o < 12 (N-tile padded to 16)
__global__ __launch_bounds__(256) void out_gemm(const float* __restrict__ hg,
                                                const float* __restrict__ W,
                                                const float* __restrict__ b,
                                                float* __restrict__ out) {
    const int lane = threadIdx.x & 31;
    const int mt = (blockIdx.x * blockDim.x + threadIdx.x) >> 5;
    const int r0 = mt * 16;
    const int node = r0 / SP;
    const int hw0 = r0 % SP;
    const int am = lane & 15;
    const int ah = lane >> 4;
    const float* fb = hg + node * CHW + hw0;
    const bool valid = (am < OUTC);
    v8f c = {};
#pragma unroll
    for (int k0 = 0; k0 < HID; k0 += 4) {
        const int ak = k0 + ah * 2;
        v2f a, bb;
        a.x = fb[ak * SP + am];
        a.y = fb[(ak + 1) * SP + am];
        bb.x = valid ? W[ak * OUTC + am] : 0.0f;
        bb.y = valid ? W[(ak + 1) * OUTC + am] : 0.0f;
        c = __builtin_amdgcn_wmma_f32_16x16x4_f32(false, a, false, bb, (short)0, c, false, false);
    }
    if (valid) {
        const float bo = b[am];
        float* ob = out + node * (OUTC * SP) + am * SP + hw0 + ah * 8;
#pragma unroll
        for (int j = 0; j < 8; ++j) ob[j] = c[j] + bo;
    }
}

// ---------------------------------------------------------------- final gate
// hg[n,c,h,w] = mean_w(hv[n,c,h,:]) * hv[n,c,h,w]   (one thread per (n,c,h))
__global__ void hg_kernel(const float* __restrict__ hv, float* __restrict__ hg) {
    int i = blockIdx.x * blockDim.x + threadIdx.x;   // 3,840,000 rows of 4
    if (i >= NODE_Q) return;
    const float4 v = *(const float4*)(hv + i * 4);
    float m = (v.x + v.y + v.z + v.w) * 0.25f;
    float4 r;
    r.x = v.x * m; r.y = v.y * m; r.z = v.z * m; r.w = v.w * m;
    *(float4*)(hg + i * 4) = r;
}

// ---------------------------------------------------------------- launcher
extern "C" void kernel_launch(void* const* d_in, const int* in_sizes, int n_in,
                              void* d_out, int out_size, void* d_ws, size_t ws_size,
                              hipStream_t stream) {
    const float* node_feats = (const float*)d_in[0];
    const float* edge_feats = (const float*)d_in[1];
    const int*   src        = (const int*)d_in[2];
    const int*   dst        = (const int*)d_in[3];
    const float* bn_gamma   = (const float*)d_in[4];
    const float* bn_beta    = (const float*)d_in[5];
    const float* gen_W      = (const float*)d_in[6];
    const float* gen_b      = (const float*)d_in[7];
    const float* out_W      = (const float*)d_in[8];
    const float* out_b      = (const float*)d_in[9];
    float* out = (float*)d_out;

    float* hv    = (float*)d_ws;
    float* hv1   = hv   + NODE_ELEMS;
    float* agg   = hv1  + NODE_ELEMS;
    float* mx    = agg  + NODE_ELEMS;
    float* ssum  = mx   + NODE_ELEMS;
    float* stats = ssum + NODE_ELEMS;     // 256 floats

    const int nodeQBlocks = (NODE_Q + 255) / 256;   // 15000
    const int edgeQBlocks = (EDGE_Q + 255) / 256;   // 150000

    copy4_kernel<<<nodeQBlocks, 256, 0, stream>>>((float4*)hv, (const float4*)node_feats, NODE_Q);

    for (int l = 0; l < 3; ++l) {
        // zero agg, mx, ssum, stats (contiguous region; count divisible by 4)
        zero4_kernel<<<4096, 256, 0, stream>>>((float4*)agg, (3 * NODE_ELEMS + 256) / 4);
        bn_stats<<<dim3(64, 20), 256, 0, stream>>>(hv, stats, stats + 64);
        bn_final<<<1, 64, 0, stream>>>(stats, bn_gamma + l * HID, bn_beta + l * HID);
        bn_apply<<<nodeQBlocks, 256, 0, stream>>>(hv, hv1, stats + 128, stats + 192);
        edge_pass1<<<edgeQBlocks, 256, 0, stream>>>(hv1, edge_feats, src, dst, mx);
        edge_pass2<<<edgeQBlocks, 256, 0, stream>>>(hv1, edge_feats, src, dst, mx, ssum, agg);
        feats_kernel<<<nodeQBlocks, 256, 0, stream>>>(agg, ssum, hv1);
        // 15000 waves / 8 per block = 1875 blocks (exact -> EXEC all 1s for WMMA)
        genconv_gemm<<<1875, 256, 0, stream>>>(agg, gen_W + l * HID * HID,
                                               gen_b + l * HID, hv);
    }

    hg_kernel<<<nodeQBlocks, 256, 0, stream>>>(hv, hv1);
    // 15000 M-tiles / 8 waves per block = 1875 blocks (exact)
    out_gemm<<<1875, 256, 0, stream>>>(hv1, out_W, out_b, out);
}